// GCN_52656299049248
// MI455X (gfx1250) — compile-verified
//
#include <hip/hip_runtime.h>

#define NND 100000
#define NED 1600000
#define CH 128
#define NG 512
#define NC 64
#define BN_EPS 1e-5f

typedef float v2f __attribute__((ext_vector_type(2)));
typedef float v8f __attribute__((ext_vector_type(8)));

#define AS_STRIDE 132  // 132 % 64 = 4 -> bank = (4*m + k) % 64, conflict-free for WMMA A reads

// ---------------------------------------------------------------------------
// GEMM: Y[nrows x 128] = X[nrows x 128] @ W[128 x 128], fp32 via V_WMMA_F32_16X16X4_F32
// Block = 256 threads = 8 waves. Block computes a 64-row x 128-col stripe.
// Wave w owns columns [16w, 16w+16); loops 4 row-subtiles of 16.
// ---------------------------------------------------------------------------
__global__ __launch_bounds__(256) void k_gemm128(const float* __restrict__ X,
                                                 const float* __restrict__ W,
                                                 float* __restrict__ Y, int nrows)
{
    __shared__ float As[64 * AS_STRIDE];
    const int tid   = threadIdx.x;
    const int mbase = blockIdx.x * 64;
    const bool full = (mbase + 64 <= nrows);  // uniform over the block

    if (full) {
        // Fast path: float4 staging -> global_load_b128 + ds_store_b128
        const float4* Xv = (const float4*)(X + (size_t)mbase * CH);
        #pragma unroll
        for (int j = 0; j < 8; ++j) {
            int i = tid + j * 256;            // 64 rows * 32 float4 = 2048
            int r = i >> 5, c4 = i & 31;
            float4 v = Xv[r * 32 + c4];
            *(float4*)&As[r * AS_STRIDE + c4 * 4] = v;
        }
    } else {
        for (int i = tid; i < 64 * CH; i += 256) {
            int r = i >> 7, c = i & 127;
            int row = mbase + r;
            As[r * AS_STRIDE + c] = (row < nrows) ? X[(size_t)row * CH + c] : 0.0f;
        }
    }
    __syncthreads();

    const int wave = tid >> 5, lane = tid & 31;
    const int hi = lane >> 4, lo = lane & 15;
    const int n0 = wave * 16;

    v8f acc[4] = {};  // 4 row-subtiles of 16x16 fp32 accumulators

    const float* wp = W + n0 + lo;  // column n = n0+lo of W (row-major [K][N])
    #pragma unroll 4
    for (int k = 0; k < CH; k += 4) {
        const int k0 = k + 2 * hi;
        // B element (K = k0+v, N = n0+lo)
        v2f b;
        b.x = wp[(size_t)k0 * CH];
        b.y = wp[(size_t)(k0 + 1) * CH];
        #pragma unroll
        for (int t = 0; t < 4; ++t) {
            // A element (M = t*16 + lo, K = k0+v), contiguous -> ds_load_b64
            const float* ap = &As[(t * 16 + lo) * AS_STRIDE + k0];
            v2f a; a.x = ap[0]; a.y = ap[1];
            acc[t] = __builtin_amdgcn_wmma_f32_16x16x4_f32(
                /*neg_a=*/false, a, /*neg_b=*/false, b,
                /*c_mod=*/(short)0, acc[t], /*reuse_a=*/false, /*reuse_b=*/false);
        }
    }

    // D layout: VGPR r -> row r + 8*hi, col lo
    if (full) {
        float* yp = Y + (size_t)mbase * CH + n0 + lo;
        #pragma unroll
        for (int t = 0; t < 4; ++t) {
            #pragma unroll
            for (int r = 0; r < 8; ++r) {
                yp[(size_t)(t * 16 + r + 8 * hi) * CH] = acc[t][r];
            }
        }
    } else {
        #pragma unroll
        for (int t = 0; t < 4; ++t) {
            #pragma unroll
            for (int r = 0; r < 8; ++r) {
                int row = mbase + t * 16 + r + 8 * hi;
                if (row < nrows) Y[(size_t)row * CH + n0 + lo] = acc[t][r];
            }
        }
    }
}

// ---------------------------------------------------------------------------
// Utility / graph kernels
// ---------------------------------------------------------------------------
__global__ void k_zero(float* __restrict__ p, long long n)
{
    long long i = (long long)blockIdx.x * blockDim.x + threadIdx.x;
    if (i < n) p[i] = 0.0f;
}

__global__ void k_degree(const int* __restrict__ dst, float* __restrict__ deg)
{
    int e = blockIdx.x * blockDim.x + threadIdx.x;
    if (e < NED) atomicAdd(&deg[dst[e]], 1.0f);
}

__global__ void k_dinv(const float* __restrict__ deg, float* __restrict__ dinv)
{
    int i = blockIdx.x * blockDim.x + threadIdx.x;
    if (i < NND) dinv[i] = rsqrtf(deg[i] + 1.0f);
}

// agg = H * dinv^2 + bias   (self-loop term; fully initializes agg before scatter)
__global__ void k_self_init(const float* __restrict__ H, const float* __restrict__ dinv,
                            const float* __restrict__ bias, float* __restrict__ agg)
{
    long long idx = (long long)blockIdx.x * blockDim.x + threadIdx.x;
    if (idx >= (long long)NND * CH) return;
    int i = (int)(idx >> 7), c = (int)(idx & 127);
    float di = dinv[i];
    agg[idx] = H[idx] * di * di + bias[c];
}

// One wave per edge; each lane handles 4 channels (float4 gather + 4 f32 atomics)
__global__ __launch_bounds__(256) void k_edge_scatter(const int* __restrict__ src,
                                                      const int* __restrict__ dst,
                                                      const float* __restrict__ dinv,
                                                      const float* __restrict__ H,
                                                      float* __restrict__ agg)
{
    long long e = (long long)blockIdx.x * 8 + (threadIdx.x >> 5);
    if (e >= NED) return;
    int lane = threadIdx.x & 31;
    int s = src[e], d = dst[e];
    float coef = dinv[s] * dinv[d];
    const float4 v = ((const float4*)(H + (size_t)s * CH))[lane];
    float* ag = agg + (size_t)d * CH + lane * 4;
    atomicAdd(ag + 0, coef * v.x);
    atomicAdd(ag + 1, coef * v.y);
    atomicAdd(ag + 2, coef * v.z);
    atomicAdd(ag + 3, coef * v.w);
}

// Per-channel sum and sum-of-squares. Launch: 512 blocks x 256 threads
// (stride 131072 is a multiple of 128, so each thread stays on one channel).
__global__ void k_bn_stats(const float* __restrict__ A, float* __restrict__ sums)
{
    long long t0 = (long long)blockIdx.x * blockDim.x + threadIdx.x;
    long long stride = (long long)gridDim.x * blockDim.x;
    int c = (int)(t0 & 127);
    float s = 0.0f, q = 0.0f;
    for (long long i = t0; i < (long long)NND * CH; i += stride) {
        float v = A[i];
        s += v; q += v * v;
    }
    atomicAdd(&sums[c], s);
    atomicAdd(&sums[CH + c], q);
}

__global__ void k_bn_finalize(const float* __restrict__ sums, const float* __restrict__ g,
                              const float* __restrict__ be, float* __restrict__ ss)
{
    int c = threadIdx.x;
    if (c >= CH) return;
    float mu  = sums[c] / (float)NND;
    float var = sums[CH + c] / (float)NND - mu * mu;
    float sc  = g[c] * rsqrtf(var + BN_EPS);
    ss[c]      = sc;               // scale
    ss[CH + c] = be[c] - mu * sc;  // shift
}

// out = relu(A*scale + shift [+ res])
__global__ void k_bn_apply(const float* __restrict__ A, const float* __restrict__ ss,
                           const float* __restrict__ res, float* __restrict__ out)
{
    long long idx = (long long)blockIdx.x * blockDim.x + threadIdx.x;
    if (idx >= (long long)NND * CH) return;
    int c = (int)(idx & 127);
    float v = A[idx] * ss[c] + ss[CH + c];
    if (res) v += res[idx];
    out[idx] = fmaxf(v, 0.0f);
}

__global__ void k_pool_cnt(const int* __restrict__ batch, float* __restrict__ cnt)
{
    int i = blockIdx.x * blockDim.x + threadIdx.x;
    if (i < NND) atomicAdd(&cnt[batch[i]], 1.0f);
}

// h >= 0 post-ReLU, so uint bit-pattern compare == float compare; mx init 0 matches
// the reference's where(cnt>0, max, 0).
__global__ void k_pool_accum(const float* __restrict__ H, const int* __restrict__ batch,
                             float* __restrict__ sum, unsigned int* __restrict__ mx)
{
    long long idx = (long long)blockIdx.x * blockDim.x + threadIdx.x;
    if (idx >= (long long)NND * CH) return;
    int i = (int)(idx >> 7), c = (int)(idx & 127);
    int g = batch[i];
    float v = H[idx];
    atomicAdd(&sum[(size_t)g * CH + c], v);
    atomicMax(&mx[(size_t)g * CH + c], __float_as_uint(v));
}

// out[g][o] = bh[o] + sum_c mean*Wh[c][o] + s*Wh[128+c][o] + mx*Wh[256+c][o]
__global__ void k_head(const float* __restrict__ sum, const unsigned int* __restrict__ mx,
                       const float* __restrict__ cnt, const float* __restrict__ Wh,
                       const float* __restrict__ bh, float* __restrict__ out)
{
    int idx = blockIdx.x * blockDim.x + threadIdx.x;
    if (idx >= NG * NC) return;
    int g = idx >> 6, o = idx & 63;
    float cn = fmaxf(cnt[g], 1.0f);
    float acc = bh[o];
    #pragma unroll 4
    for (int c = 0; c < CH; ++c) {
        float s = sum[(size_t)g * CH + c];
        float m = __uint_as_float(mx[(size_t)g * CH + c]);
        acc += (s / cn) * Wh[c * NC + o] + s * Wh[(CH + c) * NC + o] + m * Wh[(2 * CH + c) * NC + o];
    }
    out[idx] = acc;
}

// ---------------------------------------------------------------------------
// Launch
// ---------------------------------------------------------------------------
extern "C" void kernel_launch(void* const* d_in, const int* in_sizes, int n_in,
                              void* d_out, int out_size, void* d_ws, size_t ws_size,
                              hipStream_t stream)
{
    (void)in_sizes; (void)n_in; (void)out_size; (void)ws_size;

    const float* x     = (const float*)d_in[0];
    const int*   eidx  = (const int*)d_in[1];
    const int*   src   = eidx;
    const int*   dstp  = eidx + NED;
    const int*   batch = (const int*)d_in[2];
    const float* W[3]  = { (const float*)d_in[3], (const float*)d_in[7],  (const float*)d_in[11] };
    const float* bb[3] = { (const float*)d_in[4], (const float*)d_in[8],  (const float*)d_in[12] };
    const float* gg[3] = { (const float*)d_in[5], (const float*)d_in[9],  (const float*)d_in[13] };
    const float* be[3] = { (const float*)d_in[6], (const float*)d_in[10], (const float*)d_in[14] };
    const float* Wh    = (const float*)d_in[15];
    const float* bh    = (const float*)d_in[16];
    float* out = (float*)d_out;

    // Workspace partition (256B aligned)
    char* ws = (char*)d_ws;
    size_t off = 0;
    auto take = [&](size_t bytes) { char* p = ws + off; off = (off + bytes + 255) & ~(size_t)255; return p; };
    const size_t NODEFEAT = (size_t)NND * CH * sizeof(float);
    float* G     = (float*)take(NODEFEAT);   // gemm output
    float* AGG   = (float*)take(NODEFEAT);   // scatter accumulator
    float* H1    = (float*)take(NODEFEAT);   // layer outputs / residuals
    float* H2    = (float*)take(NODEFEAT);
    float* deg   = (float*)take(NND * sizeof(float));
    float* dinv  = (float*)take(NND * sizeof(float));
    float* bnsum = (float*)take(2 * CH * sizeof(float));
    float* bnss  = (float*)take(2 * CH * sizeof(float));
    float* psum  = (float*)take((size_t)NG * CH * sizeof(float));
    unsigned int* pmax = (unsigned int*)take((size_t)NG * CH * sizeof(unsigned int));
    float* pcnt  = (float*)take(NG * sizeof(float));

    const int T = 256;
    const long long NF = (long long)NND * CH;
    const int nfBlocks   = (int)((NF + T - 1) / T);
    const int gemmBlocks = (NND + 63) / 64;
    const int edgeBlocks = (NED + 7) / 8;  // 8 edges (waves) per block

    // Degrees + dinv
    k_zero<<<(NND + T - 1) / T, T, 0, stream>>>(deg, NND);
    k_degree<<<(NED + T - 1) / T, T, 0, stream>>>(dstp, deg);
    k_dinv<<<(NND + T - 1) / T, T, 0, stream>>>(deg, dinv);

    const float* layer_in[3]  = { x,  H1, H2 };
    const float* layer_res[3] = { nullptr, H1, H2 };
    float*       layer_out[3] = { H1, H2, H1 };  // H1 reusable for layer 3

    for (int l = 0; l < 3; ++l) {
        k_gemm128<<<gemmBlocks, T, 0, stream>>>(layer_in[l], W[l], G, NND);
        k_self_init<<<nfBlocks, T, 0, stream>>>(G, dinv, bb[l], AGG);
        k_edge_scatter<<<edgeBlocks, T, 0, stream>>>(src, dstp, dinv, G, AGG);
        k_zero<<<1, T, 0, stream>>>(bnsum, 2 * CH);
        k_bn_stats<<<512, T, 0, stream>>>(AGG, bnsum);
        k_bn_finalize<<<1, CH, 0, stream>>>(bnsum, gg[l], be[l], bnss);
        k_bn_apply<<<nfBlocks, T, 0, stream>>>(AGG, bnss, layer_res[l], layer_out[l]);
    }

    // Pooling (H1 holds final node features)
    const long long PF = (long long)NG * CH;
    k_zero<<<(int)((PF + T - 1) / T), T, 0, stream>>>(psum, PF);
    k_zero<<<(int)((PF + T - 1) / T), T, 0, stream>>>((float*)pmax, PF);
    k_zero<<<1, T, 0, stream>>>(pcnt, NG);
    k_pool_cnt<<<(NND + T - 1) / T, T, 0, stream>>>(batch, pcnt);
    k_pool_accum<<<nfBlocks, T, 0, stream>>>(H1, batch, psum, pmax);

    // Head
    k_head<<<(NG * NC + T - 1) / T, T, 0, stream>>>(psum, pmax, pcnt, Wh, bh, out);
}